// jarvis_71201967833879
// MI455X (gfx1250) — compile-verified
//
#include <hip/hip_runtime.h>
#include <math.h>

typedef float v2f __attribute__((ext_vector_type(2)));
typedef float v8f __attribute__((ext_vector_type(8)));

#define BATCH 8
#define NPTS  4096
#define KNN   16
#define MTOT  (BATCH * NPTS)
#define EPSV  1e-5f

// ---------------------------------------------------------------------------
// KNN (top-16 by squared distance, includes self) + centered covariance.
// One block handles 256 query points; entire batch point cloud (48 KB) lives
// in LDS (WGP has 320 KB). Insertion sort in registers, fully unrolled.
// feat[b,n,0:3] = xyz ; feat[b,n,3:12] = 3x3 covariance (row major).
// ---------------------------------------------------------------------------
__global__ __launch_bounds__(256) void knn_feat_kernel(const float* __restrict__ data,
                                                       float* __restrict__ feat) {
  __shared__ float px[NPTS], py[NPTS], pz[NPTS];
  const int b     = blockIdx.x >> 4;           // 16 blocks per batch
  const int qbase = (blockIdx.x & 15) << 8;
  const float* dp = data + (size_t)b * NPTS * 3;
  for (int i = threadIdx.x; i < NPTS; i += 256) {
    px[i] = dp[i * 3 + 0]; py[i] = dp[i * 3 + 1]; pz[i] = dp[i * 3 + 2];
  }
  __syncthreads();
  const int q = qbase + threadIdx.x;
  const float qx = px[q], qy = py[q], qz = pz[q];
  float bd[KNN]; int bi[KNN];
#pragma unroll
  for (int s = 0; s < KNN; ++s) { bd[s] = 3.4e38f; bi[s] = 0; }
  for (int j = 0; j < NPTS; ++j) {
    const float dx = px[j] - qx, dy = py[j] - qy, dz = pz[j] - qz;
    const float d = dx * dx + dy * dy + dz * dz;
    if (d < bd[KNN - 1]) {
      float dd = d; int jj = j;
#pragma unroll
      for (int s = 0; s < KNN; ++s) {
        if (dd < bd[s]) { float t = bd[s]; int ti = bi[s]; bd[s] = dd; bi[s] = jj; dd = t; jj = ti; }
      }
    }
  }
  float nx[KNN], ny[KNN], nz[KNN];
  float mx = 0.f, my = 0.f, mz = 0.f;
#pragma unroll
  for (int s = 0; s < KNN; ++s) {
    nx[s] = px[bi[s]]; ny[s] = py[bi[s]]; nz[s] = pz[bi[s]];
    mx += nx[s]; my += ny[s]; mz += nz[s];
  }
  mx *= (1.f / KNN); my *= (1.f / KNN); mz *= (1.f / KNN);
  float c00 = 0, c01 = 0, c02 = 0, c11 = 0, c12 = 0, c22 = 0;
#pragma unroll
  for (int s = 0; s < KNN; ++s) {
    const float ex = nx[s] - mx, ey = ny[s] - my, ez = nz[s] - mz;
    c00 += ex * ex; c01 += ex * ey; c02 += ex * ez;
    c11 += ey * ey; c12 += ey * ez; c22 += ez * ez;
  }
  float* f = feat + (size_t)(b * NPTS + q) * 12;
  f[0] = qx;  f[1] = qy;  f[2] = qz;
  f[3] = c00; f[4] = c01; f[5] = c02;
  f[6] = c01; f[7] = c11; f[8] = c12;
  f[9] = c02; f[10] = c12; f[11] = c22;
}

// ---------------------------------------------------------------------------
// Per-point GEMM  Y[M,Cout] = X[M,Cin] @ W[Cout,Cin]^T + bias, f32 WMMA.
// One wave per 16x16 output tile, K-loop of V_WMMA_F32_16X16X4_F32.
// Also accumulates per-channel sum / sum-of-squares for BN (global atomics).
// ---------------------------------------------------------------------------
template <bool STORE>
__global__ __launch_bounds__(256) void gemm_stats_kernel(
    const float* __restrict__ X, const float* __restrict__ W,
    const float* __restrict__ bias, float* __restrict__ Y,
    float* __restrict__ csum, float* __restrict__ csq,
    int Mtot, int Cin, int Cout) {
  const int wave = threadIdx.x >> 5;
  const int lane = threadIdx.x & 31;
  const int colTiles = Cout >> 4;
  const int tile = blockIdx.x * 8 + wave;
  if (tile >= (Mtot >> 4) * colTiles) return;
  const int rt = tile / colTiles, ct = tile % colTiles;
  const int row0 = rt << 4;
  const int lr = lane & 15;            // A row / B col within tile
  const int kh = (lane >> 4) << 1;     // K sub-offset: 0 or 2
  const float* xr = X + (size_t)(row0 + lr) * Cin;
  const float* wr = W + (size_t)(ct * 16 + lr) * Cin;
  v8f acc = {};
  for (int k0 = 0; k0 < Cin; k0 += 4) {
    v2f a, bfr;
    a.x = xr[k0 + kh];     a.y = xr[k0 + kh + 1];
    bfr.x = wr[k0 + kh];   bfr.y = wr[k0 + kh + 1];
    acc = __builtin_amdgcn_wmma_f32_16x16x4_f32(false, a, false, bfr,
                                                (short)0, acc, false, false);
  }
  const int ch = ct * 16 + lr;
  const float bv = bias[ch];
  const int half8 = (lane >> 4) << 3;  // row offset 0 or 8 in C layout
  float ps = 0.f, pq = 0.f;
#pragma unroll
  for (int v = 0; v < 8; ++v) {
    const float y = acc[v] + bv;
    if (STORE) Y[(size_t)(row0 + half8 + v) * Cout + ch] = y;
    ps += y; pq += y * y;
  }
  atomicAdd(&csum[ch], ps);
  atomicAdd(&csq[ch], pq);
}

// ---------------------------------------------------------------------------
// conv4 pass B: recompute GEMM tile, apply BN (precomputed scale/shift),
// ReLU, and atomic max-pool over N into pooled[b, ch]. Post-ReLU values are
// >= 0 so integer-bit atomicMax == float max (pooled zero-initialized).
// ---------------------------------------------------------------------------
__global__ __launch_bounds__(256) void gemm_bn_maxpool_kernel(
    const float* __restrict__ X, const float* __restrict__ W,
    const float* __restrict__ bias, const float* __restrict__ scale,
    const float* __restrict__ shift, int* __restrict__ pooled,
    int Mtot, int Cin, int Cout) {
  const int wave = threadIdx.x >> 5;
  const int lane = threadIdx.x & 31;
  const int colTiles = Cout >> 4;
  const int tile = blockIdx.x * 8 + wave;
  if (tile >= (Mtot >> 4) * colTiles) return;
  const int rt = tile / colTiles, ct = tile % colTiles;
  const int row0 = rt << 4;
  const int lr = lane & 15;
  const int kh = (lane >> 4) << 1;
  const float* xr = X + (size_t)(row0 + lr) * Cin;
  const float* wr = W + (size_t)(ct * 16 + lr) * Cin;
  v8f acc = {};
  for (int k0 = 0; k0 < Cin; k0 += 4) {
    v2f a, bfr;
    a.x = xr[k0 + kh];   a.y = xr[k0 + kh + 1];
    bfr.x = wr[k0 + kh]; bfr.y = wr[k0 + kh + 1];
    acc = __builtin_amdgcn_wmma_f32_16x16x4_f32(false, a, false, bfr,
                                                (short)0, acc, false, false);
  }
  const int ch = ct * 16 + lr;
  const float bv = bias[ch];
  const float sc = scale[ch], sh = shift[ch];
  float mxv = 0.f;
#pragma unroll
  for (int v = 0; v < 8; ++v) {
    const float y = acc[v] + bv;
    mxv = fmaxf(mxv, fmaxf(y * sc + sh, 0.f));
  }
  const int bidx = row0 / NPTS;        // 4096 % 16 == 0: tile never crosses batch
  atomicMax(&pooled[bidx * Cout + ch], __float_as_int(mxv));
}

// BN finalize: scale = g*rsqrt(var+eps), shift = beta - mean*scale
__global__ void bn_finalize_kernel(const float* __restrict__ csum,
                                   const float* __restrict__ csq,
                                   const float* __restrict__ g,
                                   const float* __restrict__ beta,
                                   float* __restrict__ scale,
                                   float* __restrict__ shift,
                                   int C, float invN) {
  const int c = blockIdx.x * blockDim.x + threadIdx.x;
  if (c >= C) return;
  const float m = csum[c] * invN;
  const float v = csq[c] * invN - m * m;
  const float sc = g[c] * rsqrtf(v + EPSV);
  scale[c] = sc;
  shift[c] = beta[c] - m * sc;
}

// elementwise BN + ReLU (in place)
__global__ __launch_bounds__(256) void bn_apply_relu_kernel(
    float* __restrict__ Y, const float* __restrict__ scale,
    const float* __restrict__ shift, int total, int C) {
  const int i = blockIdx.x * blockDim.x + threadIdx.x;
  if (i >= total) return;
  const int c = i % C;
  Y[i] = fmaxf(Y[i] * scale[c] + shift[c], 0.f);
}

// FC + BN(batch of 8) + ReLU: one thread per output channel, batch loop local.
__global__ void fc_bn_relu_kernel(const float* __restrict__ X,
                                  const float* __restrict__ W,
                                  const float* __restrict__ bias,
                                  const float* __restrict__ g,
                                  const float* __restrict__ beta,
                                  float* __restrict__ out, int Cin, int Cout) {
  const int ch = blockIdx.x * blockDim.x + threadIdx.x;
  if (ch >= Cout) return;
  float y[BATCH];
  for (int r = 0; r < BATCH; ++r) {
    float s = bias[ch];
    const float* xr = X + (size_t)r * Cin;
    const float* wr = W + (size_t)ch * Cin;
    for (int c = 0; c < Cin; ++c) s += xr[c] * wr[c];
    y[r] = s;
  }
  float m = 0.f;
#pragma unroll
  for (int r = 0; r < BATCH; ++r) m += y[r];
  m *= (1.f / BATCH);
  float v = 0.f;
#pragma unroll
  for (int r = 0; r < BATCH; ++r) { const float d = y[r] - m; v += d * d; }
  v *= (1.f / BATCH);
  const float sc = g[ch] * rsqrtf(v + EPSV);
  const float sh = beta[ch] - m * sc;
#pragma unroll
  for (int r = 0; r < BATCH; ++r) out[(size_t)r * Cout + ch] = fmaxf(y[r] * sc + sh, 0.f);
}

// fc4 (64->4, tanh) + fc5 (4->1, sigmoid): one thread per batch row.
__global__ void fc45_kernel(const float* __restrict__ X,
                            const float* __restrict__ W4, const float* __restrict__ b4,
                            const float* __restrict__ W5, const float* __restrict__ b5,
                            float* __restrict__ out) {
  const int r = threadIdx.x;
  if (r >= BATCH) return;
  float t[4];
#pragma unroll
  for (int o = 0; o < 4; ++o) {
    float s = b4[o];
    for (int c = 0; c < 64; ++c) s += X[r * 64 + c] * W4[o * 64 + c];
    t[o] = tanhf(s);
  }
  float s = b5[0];
#pragma unroll
  for (int o = 0; o < 4; ++o) s += t[o] * W5[o];
  out[r] = 1.f / (1.f + expf(-s));
}

// ---------------------------------------------------------------------------
// Workspace layout (floats)
// ---------------------------------------------------------------------------
#define OFF_FEAT 0u                       // 32768*12 = 393216
#define OFF_Y1   393216u                  // 32768*32 = 1048576
#define OFF_Y2   (OFF_Y1 + 1048576u)      // 1048576
#define OFF_ZERO (OFF_Y2 + 1048576u)      // atomically-updated region (memset 0)
//   zero region: sum1 +0(32) sq1 +32 sum2 +64 sq2 +96 sum4 +128(512) sq4 +640(512)
//                pooled +1152 (8*512=4096)  -> total 5248 floats
#define ZERO_FLOATS 5248u
#define OFF_SC   (OFF_ZERO + ZERO_FLOATS) // scale1 +0 shift1 +32 scale2 +64 shift2 +96
                                          // scale4 +128 shift4 +640 -> 1152 floats
#define OFF_FC1  (OFF_SC + 1152u)         // 8*256 = 2048
#define OFF_FC2  (OFF_FC1 + 2048u)        // 8*128 = 1024
#define OFF_FC3  (OFF_FC2 + 1024u)        // 8*64  = 512

extern "C" void kernel_launch(void* const* d_in, const int* in_sizes, int n_in,
                              void* d_out, int out_size, void* d_ws, size_t ws_size,
                              hipStream_t stream) {
  const float* data = (const float*)d_in[0];
  const float* W1 = (const float*)d_in[1];   const float* b1  = (const float*)d_in[2];
  const float* g1 = (const float*)d_in[3];   const float* be1 = (const float*)d_in[4];
  const float* W2 = (const float*)d_in[5];   const float* b2  = (const float*)d_in[6];
  const float* g2 = (const float*)d_in[7];   const float* be2 = (const float*)d_in[8];
  const float* W4 = (const float*)d_in[9];   const float* b4  = (const float*)d_in[10];
  const float* g4 = (const float*)d_in[11];  const float* be4 = (const float*)d_in[12];
  const float* Wf1 = (const float*)d_in[13]; const float* bf1  = (const float*)d_in[14];
  const float* gf1 = (const float*)d_in[15]; const float* bef1 = (const float*)d_in[16];
  const float* Wf2 = (const float*)d_in[17]; const float* bf2  = (const float*)d_in[18];
  const float* gf2 = (const float*)d_in[19]; const float* bef2 = (const float*)d_in[20];
  const float* Wf3 = (const float*)d_in[21]; const float* bf3  = (const float*)d_in[22];
  const float* gf3 = (const float*)d_in[23]; const float* bef3 = (const float*)d_in[24];
  const float* Wf4 = (const float*)d_in[25]; const float* bf4  = (const float*)d_in[26];
  const float* Wf5 = (const float*)d_in[27]; const float* bf5  = (const float*)d_in[28];

  float* ws   = (float*)d_ws;
  float* feat = ws + OFF_FEAT;
  float* y1   = ws + OFF_Y1;
  float* y2   = ws + OFF_Y2;
  float* zb   = ws + OFF_ZERO;
  float* sum1 = zb + 0;    float* sq1 = zb + 32;
  float* sum2 = zb + 64;   float* sq2 = zb + 96;
  float* sum4 = zb + 128;  float* sq4 = zb + 640;
  float* pooled = zb + 1152;                      // [8,512]
  float* sc1 = ws + OFF_SC + 0;    float* sh1 = ws + OFF_SC + 32;
  float* sc2 = ws + OFF_SC + 64;   float* sh2 = ws + OFF_SC + 96;
  float* sc4 = ws + OFF_SC + 128;  float* sh4 = ws + OFF_SC + 640;
  float* xf1 = ws + OFF_FC1;
  float* xf2 = ws + OFF_FC2;
  float* xf3 = ws + OFF_FC3;

  // zero BN accumulators + max-pool buffer (required every call: atomics)
  hipMemsetAsync(zb, 0, ZERO_FLOATS * sizeof(float), stream);

  // 1) KNN + covariance features [32768, 12]
  knn_feat_kernel<<<BATCH * (NPTS / 256), 256, 0, stream>>>(data, feat);

  // 2) conv1: 12 -> 32  (store + stats), BN, apply
  {
    const int tiles = (MTOT / 16) * (32 / 16);
    gemm_stats_kernel<true><<<(tiles + 7) / 8, 256, 0, stream>>>(
        feat, W1, b1, y1, sum1, sq1, MTOT, 12, 32);
    bn_finalize_kernel<<<1, 32, 0, stream>>>(sum1, sq1, g1, be1, sc1, sh1, 32, 1.f / MTOT);
    bn_apply_relu_kernel<<<(MTOT * 32) / 256, 256, 0, stream>>>(y1, sc1, sh1, MTOT * 32, 32);
  }
  // 3) conv2: 32 -> 32
  {
    const int tiles = (MTOT / 16) * (32 / 16);
    gemm_stats_kernel<true><<<(tiles + 7) / 8, 256, 0, stream>>>(
        y1, W2, b2, y2, sum2, sq2, MTOT, 32, 32);
    bn_finalize_kernel<<<1, 32, 0, stream>>>(sum2, sq2, g2, be2, sc2, sh2, 32, 1.f / MTOT);
    bn_apply_relu_kernel<<<(MTOT * 32) / 256, 256, 0, stream>>>(y2, sc2, sh2, MTOT * 32, 32);
  }
  // 4) conv4: 32 -> 512, two-pass (stats, then BN+ReLU+maxpool) to cap ws use
  {
    const int tiles = (MTOT / 16) * (512 / 16);
    gemm_stats_kernel<false><<<(tiles + 7) / 8, 256, 0, stream>>>(
        y2, W4, b4, nullptr, sum4, sq4, MTOT, 32, 512);
    bn_finalize_kernel<<<2, 256, 0, stream>>>(sum4, sq4, g4, be4, sc4, sh4, 512, 1.f / MTOT);
    gemm_bn_maxpool_kernel<<<(tiles + 7) / 8, 256, 0, stream>>>(
        y2, W4, b4, sc4, sh4, (int*)pooled, MTOT, 32, 512);
  }
  // 5) FC stack on [8, 512]
  fc_bn_relu_kernel<<<1, 256, 0, stream>>>(pooled, Wf1, bf1, gf1, bef1, xf1, 512, 256);
  fc_bn_relu_kernel<<<1, 128, 0, stream>>>(xf1, Wf2, bf2, gf2, bef2, xf2, 256, 128);
  fc_bn_relu_kernel<<<1, 64, 0, stream>>>(xf2, Wf3, bf3, gf3, bef3, xf3, 128, 64);
  fc45_kernel<<<1, 32, 0, stream>>>(xf3, Wf4, bf4, Wf5, bf5, (float*)d_out);
}